// DEDistMult_39608188404224
// MI455X (gfx1250) — compile-verified
//
#include <hip/hip_runtime.h>

// DE-DistMult temporal KG scoring for MI455X (gfx1250, wave32).
// Memory-bound gather kernel: ~13.3 KB gathered per triple, ~1.74 GB total,
// => ~75us floor at 23.3 TB/s. No matmul structure (per-triple trilinear dot
// over gathered rows), so the optimized path is coalesced b128 gathers +
// hardware v_sin + wave32 reduction, not WMMA.

#define KB_S_DIM   400
#define KB_T_DIM   112
#define KB_FULL    512
#define KB_T_VEC4  (KB_T_DIM / 4)   // 28 float4 per temporal row
#define KB_S_VEC4  (KB_S_DIM / 4)   // 100 float4 per entity row
#define KB_F_VEC4  (KB_FULL / 4)    // 128 float4 per relation row

__device__ __forceinline__ float4 sin_term(const float4 amp, const float4 frq,
                                           const float4 phi, const float t) {
    // args are tiny (Xavier-scale frq/phi, t in [0,1]) -> native sin is exact here
    float4 r;
    r.x = amp.x * __sinf(fmaf(frq.x, t, phi.x));
    r.y = amp.y * __sinf(fmaf(frq.y, t, phi.y));
    r.z = amp.z * __sinf(fmaf(frq.z, t, phi.z));
    r.w = amp.w * __sinf(fmaf(frq.w, t, phi.w));
    return r;
}

__global__ __launch_bounds__(256) void de_distmult_score(
    const int*   __restrict__ s_idx, const int*   __restrict__ r_idx,
    const int*   __restrict__ o_idx,
    const float* __restrict__ y_t,   const float* __restrict__ m_t,
    const float* __restrict__ d_t,
    const float* __restrict__ e_emb, const float* __restrict__ r_emb,
    const float* __restrict__ y_frq, const float* __restrict__ y_phi,
    const float* __restrict__ y_amp,
    const float* __restrict__ m_frq, const float* __restrict__ m_phi,
    const float* __restrict__ m_amp,
    const float* __restrict__ d_frq, const float* __restrict__ d_phi,
    const float* __restrict__ d_amp,
    float* __restrict__ out, const int n)
{
    const int lane = (int)(threadIdx.x & 31u);                 // wave32 on gfx1250
    const int b    = (int)(blockIdx.x * (blockDim.x >> 5) + (threadIdx.x >> 5));
    if (b >= n) return;

    const int   sb = s_idx[b];
    const int   rb = r_idx[b];
    const int   ob = o_idx[b];
    const float yv = y_t[b], mv = m_t[b], dv = d_t[b];

    // ---- Early prefetch of all 18 temporal rows (global_prefetch_b8). ----
    // 448-byte rows -> 4 cachelines each; lanes 0-3 cover the s-row, lanes
    // 4-7 the o-row, per table. Starts the L2 fetches while the static dot
    // product below executes.
    {
        const size_t sB  = (size_t)sb * (KB_T_DIM * sizeof(float));
        const size_t oB  = (size_t)ob * (KB_T_DIM * sizeof(float));
        const size_t off = (lane < 4) ? (sB + (size_t)lane * 128)
                         : (lane < 8) ? (oB + (size_t)(lane - 4) * 128)
                                      : sB;
        __builtin_prefetch((const char*)y_frq + off, 0, 3);
        __builtin_prefetch((const char*)y_phi + off, 0, 3);
        __builtin_prefetch((const char*)y_amp + off, 0, 3);
        __builtin_prefetch((const char*)m_frq + off, 0, 3);
        __builtin_prefetch((const char*)m_phi + off, 0, 3);
        __builtin_prefetch((const char*)m_amp + off, 0, 3);
        __builtin_prefetch((const char*)d_frq + off, 0, 3);
        __builtin_prefetch((const char*)d_phi + off, 0, 3);
        __builtin_prefetch((const char*)d_amp + off, 0, 3);
    }

    // Rows are 1600 / 2048 / 448 bytes: all multiples of 16 -> float4-safe.
    const float4* __restrict__ se = (const float4*)(e_emb + (size_t)sb * KB_S_DIM);
    const float4* __restrict__ oe = (const float4*)(e_emb + (size_t)ob * KB_S_DIM);
    const float4* __restrict__ re = (const float4*)(r_emb + (size_t)rb * KB_FULL);

    float acc = 0.0f;

    // ---- Static part: 400 dims = 100 float4, stride-32 across the wave ----
#pragma unroll
    for (int i = 0; i < 4; ++i) {
        const int idx = lane + 32 * i;
        if (idx < KB_S_VEC4) {
            const float4 a = se[idx];
            const float4 c = oe[idx];
            const float4 w = re[idx];
            acc = fmaf(a.x * w.x, c.x, acc);
            acc = fmaf(a.y * w.y, c.y, acc);
            acc = fmaf(a.z * w.z, c.z, acc);
            acc = fmaf(a.w * w.w, c.w, acc);
        }
    }

    // ---- Temporal part: 112 dims = 28 float4; lanes 0-27 take one each ----
    if (lane < KB_T_VEC4) {
        const size_t si = (size_t)sb * KB_T_VEC4 + (size_t)lane;
        const size_t oi = (size_t)ob * KB_T_VEC4 + (size_t)lane;
        const float4 w  = re[KB_S_VEC4 + lane];   // r_emb dims 400..511

        const float4* yF = (const float4*)y_frq; const float4* yP = (const float4*)y_phi;
        const float4* yA = (const float4*)y_amp;
        const float4* mF = (const float4*)m_frq; const float4* mP = (const float4*)m_phi;
        const float4* mA = (const float4*)m_amp;
        const float4* dF = (const float4*)d_frq; const float4* dP = (const float4*)d_phi;
        const float4* dA = (const float4*)d_amp;

        // subject temporal embedding (4 dims)
        float4 tys = sin_term(yA[si], yF[si], yP[si], yv);
        float4 tms = sin_term(mA[si], mF[si], mP[si], mv);
        float4 tds = sin_term(dA[si], dF[si], dP[si], dv);
        float4 ts;
        ts.x = tys.x + tms.x + tds.x;
        ts.y = tys.y + tms.y + tds.y;
        ts.z = tys.z + tms.z + tds.z;
        ts.w = tys.w + tms.w + tds.w;

        // object temporal embedding (4 dims)
        float4 tyo = sin_term(yA[oi], yF[oi], yP[oi], yv);
        float4 tmo = sin_term(mA[oi], mF[oi], mP[oi], mv);
        float4 tdo = sin_term(dA[oi], dF[oi], dP[oi], dv);
        float4 to;
        to.x = tyo.x + tmo.x + tdo.x;
        to.y = tyo.y + tmo.y + tdo.y;
        to.z = tyo.z + tmo.z + tdo.z;
        to.w = tyo.w + tmo.w + tdo.w;

        acc = fmaf(ts.x * w.x, to.x, acc);
        acc = fmaf(ts.y * w.y, to.y, acc);
        acc = fmaf(ts.z * w.z, to.z, acc);
        acc = fmaf(ts.w * w.w, to.w, acc);
    }

    // ---- wave32 butterfly reduction ----
#pragma unroll
    for (int off = 16; off >= 1; off >>= 1)
        acc += __shfl_xor(acc, off, 32);

    if (lane == 0) out[b] = acc;
}

extern "C" void kernel_launch(void* const* d_in, const int* in_sizes, int n_in,
                              void* d_out, int out_size, void* d_ws, size_t ws_size,
                              hipStream_t stream) {
    (void)n_in; (void)d_ws; (void)ws_size; (void)out_size;

    const int*   s     = (const int*)  d_in[0];
    const int*   r     = (const int*)  d_in[1];
    const int*   o     = (const int*)  d_in[2];
    const float* y     = (const float*)d_in[3];
    const float* m     = (const float*)d_in[4];
    const float* d     = (const float*)d_in[5];
    const float* e_emb = (const float*)d_in[6];
    const float* r_emb = (const float*)d_in[7];
    const float* y_frq = (const float*)d_in[8];
    const float* y_phi = (const float*)d_in[9];
    const float* y_amp = (const float*)d_in[10];
    const float* m_frq = (const float*)d_in[11];
    const float* m_phi = (const float*)d_in[12];
    const float* m_amp = (const float*)d_in[13];
    const float* d_frq = (const float*)d_in[14];
    const float* d_phi = (const float*)d_in[15];
    const float* d_amp = (const float*)d_in[16];
    float* out = (float*)d_out;

    const int B = in_sizes[0];
    const int wavesPerBlock = 8;           // 256 threads = 8 wave32s
    dim3 block(32 * wavesPerBlock);
    dim3 grid((B + wavesPerBlock - 1) / wavesPerBlock);

    de_distmult_score<<<grid, block, 0, stream>>>(
        s, r, o, y, m, d, e_emb, r_emb,
        y_frq, y_phi, y_amp, m_frq, m_phi, m_amp, d_frq, d_phi, d_amp,
        out, B);
}